// Parallel_16965120819322
// MI455X (gfx1250) — compile-verified
//
#include <hip/hip_runtime.h>

typedef __attribute__((ext_vector_type(2))) float v2f;
typedef __attribute__((ext_vector_type(8))) float v8f;

#define NST 4
#define CCH 64
#define BSZ 128
#define HW  1024          // H*W = 32*32
#define KEFF 320          // 4*64 zs channels + 64 x channels
#define MTOT 256          // 4*64 output channels
#define KC   32           // k-chunk staged in LDS
#define APAD 36           // LDS row stride for A (36m mod 64 distinct -> conflict-free b64)
#define BSTR 160          // LDS k-pair row stride for B (== 32 banks -> disjoint halves)

// ---------------------------------------------------------------------------
// prep_G: build effective weight G[256][320]:
//   G[i*64+o][j*64+c] = (j==i) ? W[i,i,o,c] : sum_k W[i,i,o,k]*W[j,i,k,c]
//   G[i*64+o][256+c]  = (i==0) ? W[0,0,o,c] : 0
// ---------------------------------------------------------------------------
__global__ void prep_G(const float* __restrict__ W, float* __restrict__ G) {
    int idx = blockIdx.x * blockDim.x + threadIdx.x;
    if (idx >= MTOT * KEFF) return;
    int r   = idx / KEFF;
    int col = idx - r * KEFF;
    int i = r >> 6, o = r & 63;
    float val;
    if (col < NST * CCH) {
        int j = col >> 6, c = col & 63;
        if (j == i) {
            val = W[(size_t)(i * NST + i) * CCH * CCH + o * CCH + c];
        } else {
            const float* Wd = W + (size_t)(i * NST + i) * CCH * CCH + o * CCH; // Wd[i][o][k]
            const float* Wo = W + (size_t)(j * NST + i) * CCH * CCH;           // W[j][i][k][c]
            float s = 0.f;
            #pragma unroll 8
            for (int k = 0; k < CCH; ++k) s += Wd[k] * Wo[k * CCH + c];
            val = s;
        }
    } else {
        int c = col - NST * CCH;
        val = (i == 0) ? W[(size_t)o * CCH + c] : 0.f;  // W[0,0,o,c]
    }
    G[(size_t)r * KEFF + col] = val;
}

// ---------------------------------------------------------------------------
// prep_bias: bias[i*64+o] = b[i,i,o] + sum_k W[i,i,o,k] * sum_{j!=i} b[j,i,k]
// ---------------------------------------------------------------------------
__global__ void prep_bias(const float* __restrict__ W, const float* __restrict__ bvec,
                          float* __restrict__ bias) {
    int r = threadIdx.x;            // 256 threads
    int i = r >> 6, o = r & 63;
    const float* Wd = W + (size_t)(i * NST + i) * CCH * CCH + o * CCH;
    float s = bvec[(i * NST + i) * CCH + o];
    for (int k = 0; k < CCH; ++k) {
        float boff = 0.f;
        #pragma unroll
        for (int j = 0; j < NST; ++j)
            if (j != i) boff += bvec[(j * NST + i) * CCH + k];
        s += Wd[k] * boff;
    }
    bias[r] = s;
}

// ---------------------------------------------------------------------------
// fused_gemm: per batch b, out[b] (256 x 1024) = G (256 x 320) * in[b] (320 x 1024)
// where in[b] = concat over channel dim of zs[0..3, b] and x[b].
// Grid: 128 batches * 16 nTiles * 2 mTiles; 8 wave32s per workgroup;
// each wave computes a 16(M) x 64(N) tile via v_wmma_f32_16x16x4_f32.
// B staged in LDS interleaved by k-parity so each B fragment is ONE aligned
// ds_load_b64 straight into the WMMA operand pair (no v_mov packing).
// ---------------------------------------------------------------------------
__global__ __launch_bounds__(256) void fused_gemm(
    const float* __restrict__ x, const float* __restrict__ zs,
    const float* __restrict__ G, const float* __restrict__ bias,
    float* __restrict__ out_zs) {

    __shared__ float lA[128 * APAD];       // A chunk: 128 M-rows x 32 k (padded rows)
    __shared__ float lB[(KC / 2) * BSTR];  // B chunk: 16 k-pairs x (64n x 2k) interleaved

    const int id    = blockIdx.x;
    const int mTile = id & 1;
    const int nTile = (id >> 1) & 15;
    const int b     = id >> 5;

    const int t     = threadIdx.x;
    const int wave  = t >> 5;
    const int lane  = t & 31;
    const int lhalf = lane >> 4;   // 0 for lanes 0-15, 1 for lanes 16-31
    const int lmod  = lane & 15;

    const int mBase = mTile * 128;
    const int nBase = nTile * 64;

    // cooperative-load decomposition (fixed per thread)
    const int kkA    = t & 31;        // A: k within chunk
    const int rbaseA = t >> 5;        // A: row group 0..7
    const int krowB  = t >> 3;        // B: k row 0..31
    const int colB   = (t & 7) * 8;   // B: 8-col group

    v8f acc[4] = {v8f{0,0,0,0,0,0,0,0}, v8f{0,0,0,0,0,0,0,0},
                  v8f{0,0,0,0,0,0,0,0}, v8f{0,0,0,0,0,0,0,0}};

    for (int k0 = 0; k0 < KEFF; k0 += KC) {
        // ---- cooperative load: A chunk (G strip), 4096 floats, 16 per thread
        {
            const float* gsrc = G + (size_t)(mBase + rbaseA) * KEFF + k0 + kkA;
            #pragma unroll
            for (int r0 = 0; r0 < 16; ++r0)
                lA[(rbaseA + r0 * 8) * APAD + kkA] = gsrc[(size_t)r0 * 8 * KEFF];
        }
        // ---- cooperative load: B chunk (zs/x rows), 2048 floats, 8 per thread,
        //      written interleaved: lB[(k>>1)*BSTR + 2*n + (k&1)]
        {
            const int kglob = k0 + krowB;
            const float* rp;
            if (kglob < NST * CCH) {
                int j = kglob >> 6, c = kglob & 63;
                rp = zs + ((size_t)(j * BSZ + b) * CCH + c) * HW;
            } else {
                rp = x + ((size_t)b * CCH + (kglob - NST * CCH)) * HW;
            }
            const float4* rp4 = reinterpret_cast<const float4*>(rp + nBase + colB);
            float4 v0 = rp4[0];
            float4 v1 = rp4[1];
            float* dst = &lB[(krowB >> 1) * BSTR + 2 * colB + (krowB & 1)];
            dst[0]  = v0.x; dst[2]  = v0.y; dst[4]  = v0.z; dst[6]  = v0.w;
            dst[8]  = v1.x; dst[10] = v1.y; dst[12] = v1.z; dst[14] = v1.w;
        }
        // ---- prefetch next chunk's global lines while this chunk's loads fly
        if (k0 + KC < KEFF) {
            __builtin_prefetch(G + (size_t)(mBase + rbaseA) * KEFF + (k0 + KC) + kkA, 0, 1);
            const int kg2 = k0 + KC + krowB;
            const float* rp2;
            if (kg2 < NST * CCH) {
                int j = kg2 >> 6, c = kg2 & 63;
                rp2 = zs + ((size_t)(j * BSZ + b) * CCH + c) * HW;
            } else {
                rp2 = x + ((size_t)b * CCH + (kg2 - NST * CCH)) * HW;
            }
            __builtin_prefetch(rp2 + nBase + colB, 0, 1);
        }
        __syncthreads();

        // ---- compute: 8 k-steps of 4, 4 N-subtiles each
        const int arow = wave * 16 + lmod;
        #pragma unroll
        for (int ks = 0; ks < KC; ks += 4) {
            // A 16x4 f32 layout: lane = M, VGPR0 holds K = 2*lhalf, VGPR1 K+1
            v2f a = *reinterpret_cast<const v2f*>(&lA[arow * APAD + ks + 2 * lhalf]);
            #pragma unroll
            for (int s = 0; s < 4; ++s) {
                // B 4x16 f32 layout: lane = N; (K, K+1) adjacent in LDS -> one b64
                v2f bf = *reinterpret_cast<const v2f*>(
                    &lB[((ks >> 1) + lhalf) * BSTR + 2 * (s * 16 + lmod)]);
                acc[s] = __builtin_amdgcn_wmma_f32_16x16x4_f32(
                    false, a, false, bf, (short)0, acc[s], false, false);
            }
        }
        __syncthreads();
    }

    // ---- epilogue: add bias, scatter into (N,B,C,H,W) layout
    // C/D layout: VGPR r holds row M = r + 8*lhalf, lane gives col N = lmod
    const int rowBase = mBase + wave * 16 + 8 * lhalf;
    float bv[8];
    #pragma unroll
    for (int r = 0; r < 8; ++r) bv[r] = bias[rowBase + r];

    #pragma unroll
    for (int s = 0; s < 4; ++s) {
        const int col = nBase + s * 16 + lmod;
        #pragma unroll
        for (int r = 0; r < 8; ++r) {
            const int row = rowBase + r;
            const int i = row >> 6, o = row & 63;
            out_zs[((size_t)i * (BSZ * CCH) + (size_t)b * CCH + o) * HW + col] =
                acc[s][r] + bv[r];
        }
    }
}

// ---------------------------------------------------------------------------
extern "C" void kernel_launch(void* const* d_in, const int* in_sizes, int n_in,
                              void* d_out, int out_size, void* d_ws, size_t ws_size,
                              hipStream_t stream) {
    (void)in_sizes; (void)n_in; (void)out_size; (void)ws_size;
    const float* x    = (const float*)d_in[0];   // (B,C,H,W)
    const float* zs   = (const float*)d_in[1];   // (N,B,C,H,W)
    const float* W    = (const float*)d_in[2];   // (N,N,C,C)
    const float* bvec = (const float*)d_in[3];   // (N,N,C)
    float* out = (float*)d_out;

    float* G    = (float*)d_ws;                  // 256*320 floats
    float* bias = G + (size_t)MTOT * KEFF;       // 256 floats

    const size_t xElems = (size_t)BSZ * CCH * HW;

    // output[0] = x (pass-through)
    hipMemcpyAsync(out, x, xElems * sizeof(float), hipMemcpyDeviceToDevice, stream);

    // build fused weights/bias
    prep_G<<<(MTOT * KEFF + 255) / 256, 256, 0, stream>>>(W, G);
    prep_bias<<<1, 256, 0, stream>>>(W, bvec, bias);

    // one fused GEMM produces zs_next for all 4 states
    fused_gemm<<<BSZ * 16 * 2, 256, 0, stream>>>(x, zs, G, bias, out + xElems);
}